// NeuralMemory_30236569764307
// MI455X (gfx1250) — compile-verified
//
#include <hip/hip_runtime.h>
#include <hip/hip_bf16.h>

// ---------------------------------------------------------------------------
// NeuralMemory on MI455X (gfx1250, wave32).
// Store-bandwidth-bound: 512MB grads output dominates (~25us floor at
// 23.3 TB/s). fp32 WMMA (V_WMMA_F32_16X16X4_F32) for all GEMM phases keeps
// reference precision (compute is ~0.7 GFLOP, i.e. free). The sequence cumsum
// is decomposed as a chunked scan so it doesn't serialize the device; the
// final streaming phase runs 256 workgroups with non-temporal 16B stores.
// ---------------------------------------------------------------------------

typedef float v2f __attribute__((ext_vector_type(2)));
typedef float v4f __attribute__((ext_vector_type(4)));
typedef float v8f __attribute__((ext_vector_type(8)));

#define TOK   2048      // b*n tokens
#define DDIM  128
#define NSEQ  1024
#define NB    2
#define NLAY  4
#define CHT   64        // tokens per chunk
#define NCH   16        // chunks per batch sequence
#define TD    (TOK * DDIM)

__device__ __forceinline__ float wave_reduce_sum(float v) {
    for (int o = 16; o; o >>= 1) v += __shfl_down(v, o, 32);
    return v;
}

__device__ __forceinline__ float sigmoidf_(float x) {
    return 1.0f / (1.0f + __expf(-x));
}

// ---------------------------------------------------------------------------
// c[t] = -(seq[t] . w_lr).  One wave per token.
// ---------------------------------------------------------------------------
__global__ void __launch_bounds__(256) adaptive_c_kernel(
        const float* __restrict__ seq, const float* __restrict__ w_lr,
        float* __restrict__ cbuf) {
    int t    = blockIdx.x * 8 + (threadIdx.x >> 5);
    int lane = threadIdx.x & 31;
    const v4f s4 = *(const v4f*)&seq[(size_t)t * DDIM + lane * 4];
    const v4f w4 = *(const v4f*)&w_lr[lane * 4];
    v4f m = s4 * w4;
    float s = wave_reduce_sum(m.x + m.y + m.z + m.w);
    if (lane == 0) cbuf[t] = -s;
}

// ---------------------------------------------------------------------------
// fp32 WMMA GEMM: C[M,N] = A[M,K] @ (TRANSB ? B^T : B).  TRANSB is a template
// parameter so the K-loop is branch-free.
// One wave per 16x16 output tile; K stepped by 4 via V_WMMA_F32_16X16X4_F32.
// A layout (16x4 f32): lane=row(M), VGPR0/1 = K,K+1; lanes 16-31 carry K+2,K+3.
// B layout (4x16 f32): lane=col(N), VGPR0/1 = rows K,K+1 (hi half rows K+2,K+3).
// ---------------------------------------------------------------------------
template <int TRANSB>
__global__ void __launch_bounds__(32) gemm_wmma_kernel(
        const float* __restrict__ A, const float* __restrict__ B,
        float* __restrict__ C, int M, int N, int K,
        int lda, int ldb, int ldc) {
    int ntiles = N >> 4;
    int mt = blockIdx.x / ntiles;
    int nt = blockIdx.x % ntiles;
    int lane = threadIdx.x;
    int hi   = lane >> 4;       // 0 or 1
    int id   = lane & 15;
    int m = mt * 16 + id;
    int n = nt * 16 + id;
    const float* __restrict__ Arow = A + (size_t)m * lda + 2 * hi;
    const float* __restrict__ Bbase =
        TRANSB ? (B + (size_t)n * ldb + 2 * hi)
               : (B + (size_t)(2 * hi) * ldb + n);
    v8f acc = {};
#pragma unroll 4
    for (int k = 0; k < K; k += 4) {
        v2f a, b;
        a.x = Arow[k];
        a.y = Arow[k + 1];
        if (TRANSB) {
            b.x = Bbase[k];
            b.y = Bbase[k + 1];
        } else {
            b.x = Bbase[(size_t)k * ldb];
            b.y = Bbase[(size_t)(k + 1) * ldb];
        }
        acc = __builtin_amdgcn_wmma_f32_16x16x4_f32(
                false, a, false, b, (short)0, acc, false, false);
    }
#pragma unroll
    for (int j = 0; j < 8; ++j) {
        int row = mt * 16 + j + 8 * hi;
        C[(size_t)row * ldc + nt * 16 + id] = acc[j];
    }
}

// ---------------------------------------------------------------------------
// Elementwise kernels
// ---------------------------------------------------------------------------
__global__ void __launch_bounds__(256) silu_kernel(
        const float* __restrict__ h, float* __restrict__ a, int n) {
    int i = blockIdx.x * 256 + threadIdx.x;
    if (i < n) {
        float x = h[i];
        a[i] = x * sigmoidf_(x);
    }
}

// g *= silu'(h)
__global__ void __launch_bounds__(256) dsilu_kernel(
        float* __restrict__ g, const float* __restrict__ h, int n) {
    int i = blockIdx.x * 256 + threadIdx.x;
    if (i < n) {
        float x = h[i];
        float s = sigmoidf_(x);
        g[i] *= s * (1.0f + x * (1.0f - s));
    }
}

// g3 = 2*(pred - values)/d
__global__ void __launch_bounds__(256) g3_kernel(
        const float* __restrict__ pred, const float* __restrict__ val,
        float* __restrict__ g3, int n) {
    int i = blockIdx.x * 256 + threadIdx.x;
    if (i < n) g3[i] = (pred[i] - val[i]) * (2.0f / (float)DDIM);
}

// per-token sum of squared error (one wave per token)
__global__ void __launch_bounds__(256) tokloss_kernel(
        const float* __restrict__ pred, const float* __restrict__ val,
        float* __restrict__ tokl) {
    int t    = blockIdx.x * 8 + (threadIdx.x >> 5);
    int lane = threadIdx.x & 31;
    const v4f p4 = *(const v4f*)&pred[(size_t)t * DDIM + lane * 4];
    const v4f v4 = *(const v4f*)&val[(size_t)t * DDIM + lane * 4];
    v4f e = p4 - v4;
    e = e * e;
    float s = wave_reduce_sum(e.x + e.y + e.z + e.w);
    if (lane == 0) tokl[t] = s;
}

__global__ void __launch_bounds__(256) loss_reduce_kernel(
        const float* __restrict__ tokl, float* __restrict__ out_scalar) {
    __shared__ float sm[256];
    float s = 0.0f;
    for (int i = threadIdx.x; i < TOK; i += 256) s += tokl[i];
    sm[threadIdx.x] = s;
    __syncthreads();
    for (int o = 128; o; o >>= 1) {
        if (threadIdx.x < (unsigned)o) sm[threadIdx.x] += sm[threadIdx.x + o];
        __syncthreads();
    }
    if (threadIdx.x == 0)
        *out_scalar = sm[0] * (1.0f / ((float)TOK * (float)DDIM));
}

// ---------------------------------------------------------------------------
// Phase 2a: chunk sums  S[p,q] = sum_{t in chunk} c_t * X[t,p] * G[t,q]
// WMMA GEMM with A transposed and the per-token scale folded into the A load.
// grid: (64 tiles, 128 combos) where combo = ((b*4+l)*16+ch)
// ---------------------------------------------------------------------------
__global__ void __launch_bounds__(32) chunk_sum_kernel(
        const float* __restrict__ x0, const float* __restrict__ x1,
        const float* __restrict__ x2, const float* __restrict__ x3,
        const float* __restrict__ g0, const float* __restrict__ g1,
        const float* __restrict__ g2, const float* __restrict__ g3,
        const float* __restrict__ cbuf, float* __restrict__ csum) {
    int combo = blockIdx.y;
    int b  = combo >> 6;
    int l  = (combo >> 4) & 3;
    int ch = combo & 15;
    const float* X = (l == 0) ? x0 : (l == 1) ? x1 : (l == 2) ? x2 : x3;
    const float* G = (l == 0) ? g0 : (l == 1) ? g1 : (l == 2) ? g2 : g3;
    int mt = blockIdx.x >> 3;
    int nt = blockIdx.x & 7;
    int lane = threadIdx.x;
    int hi   = lane >> 4;
    int id   = lane & 15;
    int m = mt * 16 + id;
    int n = nt * 16 + id;
    int tk0 = b * NSEQ + ch * CHT;
    const float* __restrict__ Xp = X + (size_t)(tk0 + 2 * hi) * DDIM + m;
    const float* __restrict__ Gp = G + (size_t)(tk0 + 2 * hi) * DDIM + n;
    const float* __restrict__ Cp = cbuf + tk0 + 2 * hi;
    v8f acc = {};
#pragma unroll 4
    for (int k = 0; k < CHT; k += 4) {
        v2f a, bb;
        a.x = Cp[k]     * Xp[(size_t)k * DDIM];
        a.y = Cp[k + 1] * Xp[(size_t)(k + 1) * DDIM];
        bb.x = Gp[(size_t)k * DDIM];
        bb.y = Gp[(size_t)(k + 1) * DDIM];
        acc = __builtin_amdgcn_wmma_f32_16x16x4_f32(
                false, a, false, bb, (short)0, acc, false, false);
    }
    size_t base = (size_t)combo * (DDIM * DDIM);
#pragma unroll
    for (int j = 0; j < 8; ++j)
        csum[base + (size_t)(mt * 16 + j + 8 * hi) * DDIM + nt * 16 + id] = acc[j];
}

// ---------------------------------------------------------------------------
// Phase 2b: in-place exclusive scan of the 16 chunk sums per (b,layer).
// Each thread owns one of the 16384 matrix elements.
// ---------------------------------------------------------------------------
__global__ void __launch_bounds__(256) scan_chunks_kernel(float* __restrict__ csum) {
    int combo2 = blockIdx.x >> 6;                       // (b*4+l) in 0..7
    int e = ((blockIdx.x & 63) << 8) + threadIdx.x;     // 0..16383
    size_t base = (size_t)combo2 * NCH * (DDIM * DDIM) + e;
    float run = 0.0f;
    for (int ch = 0; ch < NCH; ++ch) {
        size_t idx = base + (size_t)ch * (DDIM * DDIM);
        float v = csum[idx];
        csum[idx] = run;
        run += v;
    }
}

// ---------------------------------------------------------------------------
// Phase 2c (dominant): grid (2 column-halves, 128 (b,l,chunk) combos) = 256
// workgroups so the streaming phase fills the device. Chunk factors staged in
// LDS via b128 loads/stores (48KB/WG); each thread keeps an 8x4 f32 running-
// state tile in VGPRs, does the rank-1 update per token and streams coalesced
// non-temporal 16B stores (512MB total at line rate; NT keeps the write-once
// output out of L2). Chunk 15 also emits next_memories = W + final_state.
// ---------------------------------------------------------------------------
__global__ void __launch_bounds__(256) cumsum_stream_kernel(
        const float* __restrict__ x0, const float* __restrict__ x1,
        const float* __restrict__ x2, const float* __restrict__ x3,
        const float* __restrict__ g0, const float* __restrict__ g1,
        const float* __restrict__ g2, const float* __restrict__ g3,
        const float* __restrict__ cbuf, const float* __restrict__ prefix,
        const float* __restrict__ W,
        float* __restrict__ grads, float* __restrict__ nextmem) {
    __shared__ v4f uu4[CHT * DDIM / 4];      // 32KB: c_t * x_l[t], all 128 rows
    __shared__ v4f gg4[CHT * (DDIM / 2) / 4]; // 16KB: g_l[t], this column half
    int colhalf = blockIdx.x;                // 0 or 1
    int combo   = blockIdx.y;                // ((b*4+l)*16+ch)
    int b  = combo >> 6;
    int l  = (combo >> 4) & 3;
    int ch = combo & 15;
    const float* X = (l == 0) ? x0 : (l == 1) ? x1 : (l == 2) ? x2 : x3;
    const float* G = (l == 0) ? g0 : (l == 1) ? g1 : (l == 2) ? g2 : g3;
    int tid  = threadIdx.x;
    int tk0  = b * NSEQ + ch * CHT;
    int cbase = colhalf * (DDIM / 2);        // global column offset

    // stage u = c_t * x_l[t] (full rows), 2048 v4f
    for (int i = tid; i < CHT * DDIM / 4; i += 256) {
        int t = i >> 5;                      // (i*4)/128
        int r = (i & 31) * 4;
        int tk = tk0 + t;
        uu4[i] = cbuf[tk] * *(const v4f*)&X[(size_t)tk * DDIM + r];
    }
    // stage g_l[t] for this column half, 1024 v4f
    for (int i = tid; i < CHT * (DDIM / 2) / 4; i += 256) {
        int t = i >> 4;                      // (i*4)/64
        int c = (i & 15) * 4 + cbase;
        gg4[i] = *(const v4f*)&G[(size_t)(tk0 + t) * DDIM + c];
    }

    int cg = tid & 15;       // column group within half (4 cols each)
    int rg = tid >> 4;       // row group (8 rows each), 16 groups
    int c0 = cbase + cg * 4;
    int r0 = rg * 8;

    v4f s[8];
    size_t pbase = (size_t)combo * (DDIM * DDIM) + (size_t)r0 * DDIM + c0;
#pragma unroll
    for (int i = 0; i < 8; ++i)
        s[i] = *(const v4f*)&prefix[pbase + (size_t)i * DDIM];

    __syncthreads();

    for (int t = 0; t < CHT; ++t) {
        const v4f gv  = gg4[t * 16 + cg];
        const v4f ua0 = uu4[t * 32 + rg * 2];
        const v4f ua1 = uu4[t * 32 + rg * 2 + 1];
        size_t obase = (((size_t)(b * NSEQ + ch * CHT + t) * NLAY + l)
                        * (DDIM * DDIM)) + (size_t)r0 * DDIM + c0;
#pragma unroll
        for (int i = 0; i < 8; ++i) {
            float ui = (i < 4) ? ua0[i] : ua1[i - 4];
            s[i] += ui * gv;
            __builtin_nontemporal_store(s[i], (v4f*)&grads[obase + (size_t)i * DDIM]);
        }
    }

    if (ch == NCH - 1) {
        size_t wb = (size_t)l * (DDIM * DDIM) + (size_t)r0 * DDIM + c0;
        size_t nb = (size_t)(b * NLAY + l) * (DDIM * DDIM) + (size_t)r0 * DDIM + c0;
#pragma unroll
        for (int i = 0; i < 8; ++i) {
            const v4f w4 = *(const v4f*)&W[wb + (size_t)i * DDIM];
            *(v4f*)&nextmem[nb + (size_t)i * DDIM] = s[i] + w4;
        }
    }
}

// ---------------------------------------------------------------------------
extern "C" void kernel_launch(void* const* d_in, const int* in_sizes, int n_in,
                              void* d_out, int out_size, void* d_ws, size_t ws_size,
                              hipStream_t stream) {
    (void)in_sizes; (void)n_in; (void)out_size; (void)ws_size;

    const float* seq  = (const float*)d_in[0];
    const float* W    = (const float*)d_in[1];
    const float* Wkv  = (const float*)d_in[2];
    const float* w_lr = (const float*)d_in[3];
    // d_in[4] (w_mom) and d_in[5] (w_decay) are dead code in the reference.

    float* out = (float*)d_out;
    const size_t GRADS_ELEMS = (size_t)NB * NSEQ * NLAY * DDIM * DDIM; // 134217728
    const size_t NM_OFF   = GRADS_ELEMS;
    const size_t LOSS_OFF = NM_OFF + (size_t)NB * NLAY * DDIM * DDIM;

    // workspace carve-up (~21 MB)
    float* p = (float*)d_ws;
    float* keys   = p; p += TD;
    float* values = p; p += TD;
    float* h0 = p; p += TD;
    float* h1 = p; p += TD;
    float* h2 = p; p += TD;
    float* a1 = p; p += TD;
    float* a2 = p; p += TD;
    float* a3 = p; p += TD;
    float* pred = p; p += TD;
    float* g0 = p; p += TD;
    float* g1 = p; p += TD;
    float* g2 = p; p += TD;
    float* g3 = p; p += TD;
    float* cbuf = p; p += TOK;
    float* tokl = p; p += TOK;
    float* csum = p; p += (size_t)NB * NLAY * NCH * DDIM * DDIM; // 2M floats

    const int GEMM_TILES = (TOK / 16) * (DDIM / 16); // 1024
    const int EW_BLOCKS  = TD / 256;                 // 1024

    // ---- phase 1: per-token factors -------------------------------------
    adaptive_c_kernel<<<TOK / 8, 256, 0, stream>>>(seq, w_lr, cbuf);

    // keys / values = seq @ Wkv[:, :128] / Wkv[:, 128:]
    gemm_wmma_kernel<0><<<GEMM_TILES, 32, 0, stream>>>(
        seq, Wkv,        keys,   TOK, DDIM, DDIM, DDIM, 2 * DDIM, DDIM);
    gemm_wmma_kernel<0><<<GEMM_TILES, 32, 0, stream>>>(
        seq, Wkv + DDIM, values, TOK, DDIM, DDIM, DDIM, 2 * DDIM, DDIM);

    // forward MLP
    gemm_wmma_kernel<0><<<GEMM_TILES, 32, 0, stream>>>(
        keys, W + 0 * DDIM * DDIM, h0, TOK, DDIM, DDIM, DDIM, DDIM, DDIM);
    silu_kernel<<<EW_BLOCKS, 256, 0, stream>>>(h0, a1, TD);
    gemm_wmma_kernel<0><<<GEMM_TILES, 32, 0, stream>>>(
        a1, W + 1 * DDIM * DDIM, h1, TOK, DDIM, DDIM, DDIM, DDIM, DDIM);
    silu_kernel<<<EW_BLOCKS, 256, 0, stream>>>(h1, a2, TD);
    gemm_wmma_kernel<0><<<GEMM_TILES, 32, 0, stream>>>(
        a2, W + 2 * DDIM * DDIM, h2, TOK, DDIM, DDIM, DDIM, DDIM, DDIM);
    silu_kernel<<<EW_BLOCKS, 256, 0, stream>>>(h2, a3, TD);
    gemm_wmma_kernel<0><<<GEMM_TILES, 32, 0, stream>>>(
        a3, W + 3 * DDIM * DDIM, pred, TOK, DDIM, DDIM, DDIM, DDIM, DDIM);

    // loss head + backward deltas
    g3_kernel<<<EW_BLOCKS, 256, 0, stream>>>(pred, values, g3, TD);
    tokloss_kernel<<<TOK / 8, 256, 0, stream>>>(pred, values, tokl);
    loss_reduce_kernel<<<1, 256, 0, stream>>>(tokl, out + LOSS_OFF);

    gemm_wmma_kernel<1><<<GEMM_TILES, 32, 0, stream>>>(
        g3, W + 3 * DDIM * DDIM, g2, TOK, DDIM, DDIM, DDIM, DDIM, DDIM);
    dsilu_kernel<<<EW_BLOCKS, 256, 0, stream>>>(g2, h2, TD);
    gemm_wmma_kernel<1><<<GEMM_TILES, 32, 0, stream>>>(
        g2, W + 2 * DDIM * DDIM, g1, TOK, DDIM, DDIM, DDIM, DDIM, DDIM);
    dsilu_kernel<<<EW_BLOCKS, 256, 0, stream>>>(g1, h1, TD);
    gemm_wmma_kernel<1><<<GEMM_TILES, 32, 0, stream>>>(
        g1, W + 1 * DDIM * DDIM, g0, TOK, DDIM, DDIM, DDIM, DDIM, DDIM);
    dsilu_kernel<<<EW_BLOCKS, 256, 0, stream>>>(g0, h0, TD);

    // ---- phase 2: chunked scan of rank-1 updates ------------------------
    chunk_sum_kernel<<<dim3(64, NB * NLAY * NCH), 32, 0, stream>>>(
        keys, a1, a2, a3, g0, g1, g2, g3, cbuf, csum);
    scan_chunks_kernel<<<NB * NLAY * 64, 256, 0, stream>>>(csum);
    cumsum_stream_kernel<<<dim3(2, NB * NLAY * NCH), 256, 0, stream>>>(
        keys, a1, a2, a3, g0, g1, g2, g3, cbuf, csum, W,
        out, out + NM_OFF);
}